// RecurrentGCN_6579889897493
// MI455X (gfx1250) — compile-verified
//
#include <hip/hip_runtime.h>

// Problem constants (match reference)
#define Bk    64
#define Nn    5000
#define DINc  64
#define DMIDc 128
#define DOUTc 32
#define Ee    80000
#define Mrows (Bk * Nn)   // 320000, multiple of 64

typedef float v2f __attribute__((ext_vector_type(2)));
typedef float v8f __attribute__((ext_vector_type(8)));

__device__ __forceinline__ v8f wmma4(v2f a, v2f b, v8f c) {
  // D = A(16x4, f32) * B(4x16, f32) + C(16x16, f32)
  return __builtin_amdgcn_wmma_f32_16x16x4_f32(
      /*neg_a=*/false, a, /*neg_b=*/false, b,
      /*c_mod=*/(short)0, c, /*reuse_a=*/false, /*reuse_b=*/false);
}

__device__ __forceinline__ float sigmoidf_(float x) {
  return 1.0f / (1.0f + __expf(-x));
}

// ---------------- degree / normalization ----------------
__global__ void deg_init_kernel(float* deg) {
  int i = blockIdx.x * blockDim.x + threadIdx.x;
  if (i < Nn) deg[i] = 1.0f;                 // self-loop contribution
}

__global__ void deg_acc_kernel(const int* __restrict__ ei, float* deg) {
  int e = blockIdx.x * blockDim.x + threadIdx.x;
  if (e < Ee) atomicAdd(&deg[ei[Ee + e]], 1.0f);   // dst = edge_index[1]
}

__global__ void deg_rsqrt_kernel(float* deg) {
  int i = blockIdx.x * blockDim.x + threadIdx.x;
  if (i < Nn) deg[i] = rsqrtf(deg[i]);       // deg >= 1 always
}

// ---------------- weight swizzle: W[K x 128] -> float4{W[4k..4k+3][col]} ----------------
__global__ void swizzle_w_kernel(const float* __restrict__ W, float* __restrict__ Wsw, int K) {
  int i = blockIdx.x * blockDim.x + threadIdx.x;   // one float4 per (kb4, col)
  int total = (K >> 2) * DMIDc;
  if (i >= total) return;
  int kb4 = i / DMIDc, col = i - kb4 * DMIDc;
  float4 v;
  v.x = W[(kb4 * 4 + 0) * DMIDc + col];
  v.y = W[(kb4 * 4 + 1) * DMIDc + col];
  v.z = W[(kb4 * 4 + 2) * DMIDc + col];
  v.w = W[(kb4 * 4 + 3) * DMIDc + col];
  ((float4*)Wsw)[i] = v;
}

// ---------------- feature aggregation: agg = A_norm @ x ----------------
__global__ void agg_self_kernel(const float* __restrict__ x,
                                const float* __restrict__ dis,
                                float* __restrict__ agg) {
  size_t i = (size_t)blockIdx.x * blockDim.x + threadIdx.x;  // one float4 each
  size_t total = (size_t)Mrows * (DINc / 4);
  if (i >= total) return;
  size_t row = i >> 4;                  // b*N + n   (DIN/4 == 16)
  int n = (int)(row % Nn);
  float s = dis[n];
  s = s * s;
  float4 v = ((const float4*)x)[i];
  v.x *= s; v.y *= s; v.z *= s; v.w *= s;
  ((float4*)agg)[i] = v;
}

// Edge scatter: one wave per (edge,batch), 2 features per lane.
__global__ void agg_edges_kernel(const float* __restrict__ x,
                                 const int* __restrict__ ei,
                                 const float* __restrict__ dis,
                                 float* __restrict__ agg) {
  long long gt = (long long)blockIdx.x * blockDim.x + threadIdx.x;
  long long gw = gt >> 5;
  int lane = threadIdx.x & 31;
  if (gw >= (long long)Ee * Bk) return;
  int e = (int)(gw % Ee);
  int b = (int)(gw / Ee);                 // consecutive waves -> same batch slice (L2 friendly)
  int src = ei[e];
  int dst = ei[Ee + e];
  float w = dis[src] * dis[dst];
  const float* xr = x + ((size_t)b * Nn + src) * DINc;
  float* ar = agg + ((size_t)b * Nn + dst) * DINc;
  atomicAdd(&ar[lane],      w * xr[lane]);
  atomicAdd(&ar[lane + 32], w * xr[lane + 32]);
}

// ---------------- fused GCN-GRU cell ----------------
// Per WG: 64-row x 128-col tile, 8 waves, each wave a 2x2 block of 16x16 WMMA subtiles:
//   wave -> mp = wave&1 (msubs 2mp,2mp+1), np = wave>>1 (nsubs 2np,2np+1)
// Per k-step: 2 LDS A-frags (b64) + 2 global B-frags (b64, pre-swizzled) -> 4 WMMAs.
#define SA_STRIDE 68
#define SX_STRIDE 132
#define SA_FLOATS (64 * SA_STRIDE)
#define SX_FLOATS (64 * SX_STRIDE)
#define SMEM_FLOATS (SA_FLOATS + 3 * SX_FLOATS)

__device__ __forceinline__ void gemm_acc2(const float* __restrict__ sSrc, int lstride,
                                          const float* __restrict__ Wsw, int kcount,
                                          v8f acc[2][2],
                                          int arow0, int kh, int col0) {
  for (int kb = 0; kb < kcount; kb += 4) {
    v2f a0 = *(const v2f*)&sSrc[arow0 * lstride + kb + kh];
    v2f a1 = *(const v2f*)&sSrc[(arow0 + 16) * lstride + kb + kh];
    const float* wb = Wsw + (size_t)(kb >> 2) * (DMIDc * 4) + kh;
    v2f b0 = *(const v2f*)&wb[col0 * 4];
    v2f b1 = *(const v2f*)&wb[(col0 + 16) * 4];
    acc[0][0] = wmma4(a0, b0, acc[0][0]);
    acc[0][1] = wmma4(a0, b1, acc[0][1]);
    acc[1][0] = wmma4(a1, b0, acc[1][0]);
    acc[1][1] = wmma4(a1, b1, acc[1][1]);
  }
}

__device__ __forceinline__ void conv_stage(const float* __restrict__ sA, float* __restrict__ sC,
                                           const float* __restrict__ WgSw, const float* __restrict__ bg,
                                           int arow0, int kh, int col0, int drow0) {
  v8f acc[2][2] = {};
  gemm_acc2(sA, SA_STRIDE, WgSw, DINc, acc, arow0, kh, col0);
#pragma unroll
  for (int mi = 0; mi < 2; ++mi) {
    int drow = drow0 + mi * 16;
#pragma unroll
    for (int nj = 0; nj < 2; ++nj) {
      int col = col0 + nj * 16;
      float bias = bg[col];
#pragma unroll
      for (int r = 0; r < 8; ++r)
        sC[(drow + r) * SX_STRIDE + col] = acc[mi][nj][r] + bias;
    }
  }
}

__launch_bounds__(256)
__global__ void gru_fused_kernel(const float* __restrict__ agg,
                                 const float* __restrict__ Hprev,
                                 const float* __restrict__ WgzSw, const float* __restrict__ bg_z,
                                 const float* __restrict__ WgrSw, const float* __restrict__ bg_r,
                                 const float* __restrict__ WghSw, const float* __restrict__ bg_h,
                                 const float* __restrict__ WlzSw, const float* __restrict__ bl_z,
                                 const float* __restrict__ WlrSw, const float* __restrict__ bl_r,
                                 const float* __restrict__ WlhSw, const float* __restrict__ bl_h,
                                 float* __restrict__ Hnew) {
  extern __shared__ float smem[];
  float* sA  = smem;                    // 64 x 68  : aggregated x tile (K=64)
  float* sH  = sA + SA_FLOATS;          // 64 x 132 : H tile
  float* sC  = sH + SX_FLOATS;          // 64 x 132 : per-gate conv tile
  float* sHR = sC + SX_FLOATS;          // 64 x 132 : H*R tile

  const int tid  = threadIdx.x;
  const int lane = tid & 31;
  const int wave = tid >> 5;
  const int m0   = blockIdx.x * 64;

  // cooperative global -> LDS loads (float4, padded strides keep 16B alignment)
  for (int i = tid; i < 64 * (DINc / 4); i += 256) {
    int r = i >> 4, q = (i & 15) << 2;
    *(float4*)&sA[r * SA_STRIDE + q] = *(const float4*)&agg[(size_t)(m0 + r) * DINc + q];
  }
  for (int i = tid; i < 64 * (DMIDc / 4); i += 256) {
    int r = i >> 5, q = (i & 31) << 2;
    *(float4*)&sH[r * SX_STRIDE + q] = *(const float4*)&Hprev[(size_t)(m0 + r) * DMIDc + q];
  }
  __syncthreads();

  const int mp    = wave & 1;                   // msubs 2mp, 2mp+1
  const int np    = wave >> 1;                  // nsubs 2np, 2np+1
  const int arow0 = (mp * 2) * 16 + (lane & 15);
  const int kh    = (lane >> 4) << 1;           // 0 or 2
  const int col0  = (np * 2) * 16 + (lane & 15);
  const int drow0 = (mp * 2) * 16 + ((lane >> 4) << 3);

  v8f Z[2][2];

  // ---------- gate Z ----------
  conv_stage(sA, sC, WgzSw, bg_z, arow0, kh, col0, drow0);
  __syncthreads();
  {
    v8f acc[2][2] = {};
    gemm_acc2(sC, SX_STRIDE, WlzSw,                     DMIDc, acc, arow0, kh, col0);
    gemm_acc2(sH, SX_STRIDE, WlzSw + (DMIDc / 4) * (DMIDc * 4), DMIDc, acc, arow0, kh, col0);
#pragma unroll
    for (int mi = 0; mi < 2; ++mi)
#pragma unroll
      for (int nj = 0; nj < 2; ++nj) {
        float bias = bl_z[col0 + nj * 16];
#pragma unroll
        for (int r = 0; r < 8; ++r) Z[mi][nj][r] = sigmoidf_(acc[mi][nj][r] + bias);
      }
  }
  __syncthreads();

  // ---------- gate R -> HR = H * R ----------
  conv_stage(sA, sC, WgrSw, bg_r, arow0, kh, col0, drow0);
  __syncthreads();
  {
    v8f acc[2][2] = {};
    gemm_acc2(sC, SX_STRIDE, WlrSw,                     DMIDc, acc, arow0, kh, col0);
    gemm_acc2(sH, SX_STRIDE, WlrSw + (DMIDc / 4) * (DMIDc * 4), DMIDc, acc, arow0, kh, col0);
#pragma unroll
    for (int mi = 0; mi < 2; ++mi) {
      int drow = drow0 + mi * 16;
#pragma unroll
      for (int nj = 0; nj < 2; ++nj) {
        int col = col0 + nj * 16;
        float bias = bl_r[col];
#pragma unroll
        for (int r = 0; r < 8; ++r) {
          float rr = sigmoidf_(acc[mi][nj][r] + bias);
          sHR[(drow + r) * SX_STRIDE + col] = sH[(drow + r) * SX_STRIDE + col] * rr;
        }
      }
    }
  }
  __syncthreads();

  // ---------- candidate Ht + state update ----------
  conv_stage(sA, sC, WghSw, bg_h, arow0, kh, col0, drow0);
  __syncthreads();
  {
    v8f acc[2][2] = {};
    gemm_acc2(sC,  SX_STRIDE, WlhSw,                     DMIDc, acc, arow0, kh, col0);
    gemm_acc2(sHR, SX_STRIDE, WlhSw + (DMIDc / 4) * (DMIDc * 4), DMIDc, acc, arow0, kh, col0);
#pragma unroll
    for (int mi = 0; mi < 2; ++mi) {
      int drow = drow0 + mi * 16;
#pragma unroll
      for (int nj = 0; nj < 2; ++nj) {
        int col = col0 + nj * 16;
        float bias = bl_h[col];
#pragma unroll
        for (int r = 0; r < 8; ++r) {
          float ht = tanhf(acc[mi][nj][r] + bias);
          float h  = sH[(drow + r) * SX_STRIDE + col];
          float z  = Z[mi][nj][r];
          Hnew[(size_t)(m0 + drow + r) * DMIDc + col] = z * h + (1.0f - z) * ht;
        }
      }
    }
  }
}

// ---------------- mean over nodes + ReLU + output linear ----------------
__global__ void mean_out_kernel(const float* __restrict__ Hnew,
                                const float* __restrict__ W_out,
                                const float* __restrict__ b_out,
                                float* __restrict__ y) {
  int b = blockIdx.x;
  int d = threadIdx.x;  // 128 threads
  const float* p = Hnew + (size_t)b * Nn * DMIDc + d;
  float s = 0.0f;
#pragma unroll 4
  for (int n = 0; n < Nn; ++n) s += p[(size_t)n * DMIDc];
  __shared__ float m[DMIDc];
  float mu = s * (1.0f / (float)Nn);
  m[d] = mu > 0.0f ? mu : 0.0f;
  __syncthreads();
  if (d < DOUTc) {
    float acc = b_out[d];
#pragma unroll 4
    for (int k = 0; k < DMIDc; ++k) acc += m[k] * W_out[k * DOUTc + d];
    y[b * DOUTc + d] = acc;
  }
}

extern "C" void kernel_launch(void* const* d_in, const int* in_sizes, int n_in,
                              void* d_out, int out_size, void* d_ws, size_t ws_size,
                              hipStream_t stream) {
  (void)in_sizes; (void)n_in; (void)out_size; (void)ws_size;
  const float* x    = (const float*)d_in[0];
  const int*   ei   = (const int*)d_in[1];
  const float* H    = (const float*)d_in[2];
  const float* Wg_z = (const float*)d_in[3];  const float* bg_z = (const float*)d_in[4];
  const float* Wg_r = (const float*)d_in[5];  const float* bg_r = (const float*)d_in[6];
  const float* Wg_h = (const float*)d_in[7];  const float* bg_h = (const float*)d_in[8];
  const float* Wl_z = (const float*)d_in[9];  const float* bl_z = (const float*)d_in[10];
  const float* Wl_r = (const float*)d_in[11]; const float* bl_r = (const float*)d_in[12];
  const float* Wl_h = (const float*)d_in[13]; const float* bl_h = (const float*)d_in[14];
  const float* W_out = (const float*)d_in[15]; const float* b_out = (const float*)d_in[16];

  float* y    = (float*)d_out;                  // [B, DOUT]
  float* Hnew = y + (size_t)Bk * DOUTc;         // [B, N, DMID]

  // workspace layout:
  //   dis[N]                       @ 0
  //   swizzled weights (480 KB)    @ 32768
  //   agg[M x DIN] (81.92 MB)      @ 524288
  float* dis = (float*)d_ws;
  float* wsw = (float*)((char*)d_ws + 32768);
  float* agg = (float*)((char*)d_ws + 524288);

  float* WgzSw = wsw;                         // (64/4)*128 float4 = 8192 floats each
  float* WgrSw = WgzSw + 8192;
  float* WghSw = WgrSw + 8192;
  float* WlzSw = WghSw + 8192;                // (256/4)*128 float4 = 32768 floats each
  float* WlrSw = WlzSw + 32768;
  float* WlhSw = WlrSw + 32768;

  deg_init_kernel<<<(Nn + 255) / 256, 256, 0, stream>>>(dis);
  deg_acc_kernel<<<(Ee + 255) / 256, 256, 0, stream>>>(ei, dis);
  deg_rsqrt_kernel<<<(Nn + 255) / 256, 256, 0, stream>>>(dis);

  swizzle_w_kernel<<<((64 / 4) * DMIDc + 255) / 256, 256, 0, stream>>>(Wg_z, WgzSw, 64);
  swizzle_w_kernel<<<((64 / 4) * DMIDc + 255) / 256, 256, 0, stream>>>(Wg_r, WgrSw, 64);
  swizzle_w_kernel<<<((64 / 4) * DMIDc + 255) / 256, 256, 0, stream>>>(Wg_h, WghSw, 64);
  swizzle_w_kernel<<<((256 / 4) * DMIDc + 255) / 256, 256, 0, stream>>>(Wl_z, WlzSw, 256);
  swizzle_w_kernel<<<((256 / 4) * DMIDc + 255) / 256, 256, 0, stream>>>(Wl_r, WlrSw, 256);
  swizzle_w_kernel<<<((256 / 4) * DMIDc + 255) / 256, 256, 0, stream>>>(Wl_h, WlhSw, 256);

  {
    long long t = (long long)Mrows * (DINc / 4);
    agg_self_kernel<<<(unsigned)((t + 255) / 256), 256, 0, stream>>>(x, dis, agg);
  }
  {
    long long t = (long long)Ee * Bk * 32;
    agg_edges_kernel<<<(unsigned)((t + 255) / 256), 256, 0, stream>>>(x, ei, dis, agg);
  }

  gru_fused_kernel<<<Mrows / 64, 256, SMEM_FLOATS * sizeof(float), stream>>>(
      agg, H, WgzSw, bg_z, WgrSw, bg_r, WghSw, bg_h,
      WlzSw, bl_z, WlrSw, bl_r, WlhSw, bl_h, Hnew);

  mean_out_kernel<<<Bk, DMIDc, 0, stream>>>(Hnew, W_out, b_out, y);
}